// KDAPolicyNetwork_39831526703221
// MI455X (gfx1250) — compile-verified
//
#include <hip/hip_runtime.h>
#include <hip/hip_bf16.h>

// ---------------------------------------------------------------------------
// Problem constants (from reference): B=128, T=4096, DK=32, DV=6, E=48
// Chunked parallel scan: chunk length CH=128 -> NC=32 chunks per batch.
// Workspace layout (floats):
//   P      : B*NC*32*32 = 4,194,304   (16 MB)  chunk transition matrices
//   U      : B*NC*32*16 = 2,097,152   ( 8 MB)  chunk affine terms (DV padded->16)
//   States : B*NC*32*16 = 2,097,152   ( 8 MB)  chunk-start states
// Total ~33.6 MB of d_ws.
// ---------------------------------------------------------------------------
#define BB   128
#define TT   4096
#define DK   32
#define DV   6
#define EE   48
#define CH   128
#define NC   (TT / CH)     // 32

typedef __attribute__((ext_vector_type(2))) float v2f;
typedef __attribute__((ext_vector_type(8))) float v8f;
typedef __attribute__((ext_vector_type(4))) int  v4i;

#if defined(__gfx1250__) &&                                              \
    __has_builtin(__builtin_amdgcn_global_load_async_to_lds_b128) &&     \
    __has_builtin(__builtin_amdgcn_s_wait_asynccnt)
#define USE_ASYNC_LDS 1
#else
#define USE_ASYNC_LDS 0
#endif

#if USE_ASYNC_LDS
typedef __attribute__((address_space(1))) v4i as1_v4i;  // global (printed __device__)
typedef __attribute__((address_space(3))) v4i as3_v4i;  // LDS
__device__ __forceinline__ as1_v4i* as_global(const void* p) {
  return (as1_v4i*)(uintptr_t)p;
}
__device__ __forceinline__ as3_v4i* as_lds(void* p) {
  // Generic LDS pointers carry the LDS byte offset in the low 32 bits
  // (ISA 10.2 aperture rules).
  return (as3_v4i*)(uint32_t)(uintptr_t)p;
}
// One 16-byte lane-granular async copy: ASYNCcnt-tracked, no VGPR staging.
__device__ __forceinline__ void async_cp_b128(const void* gsrc, void* ldst) {
  __builtin_amdgcn_global_load_async_to_lds_b128(as_global(gsrc), as_lds(ldst),
                                                 0, 0);
}
#endif

__device__ __forceinline__ float lane_bcast(float v, int srclane) {
  // v_readlane_b32 -> SGPR broadcast (wave32)
  return __uint_as_float(__builtin_amdgcn_readlane(__float_as_uint(v), srclane));
}

// ---------------------------------------------------------------------------
// Phase 1: per (batch, chunk) build P (32x32 row-major) and U (32x16 row-major,
// cols >= DV are zero). One wave per chunk; lane j owns column j.
// M_t = (I - b k k^T) Diag(a);  P <- M_t P ;  U <- M_t U + b k v^T
// ---------------------------------------------------------------------------
__global__ __launch_bounds__(32) void kda_phase1(
    const float* __restrict__ kq, const float* __restrict__ v,
    const float* __restrict__ alpha, const float* __restrict__ beta,
    float* __restrict__ wsP, float* __restrict__ wsU) {
  __shared__ float Kl[CH * DK];
  __shared__ float Al[CH * DK];
  __shared__ float Vl[CH * DV];
  __shared__ float Bl[CH];

  const int lane = threadIdx.x;
  const int b = blockIdx.x / NC;
  const int c = blockIdx.x % NC;
  const size_t t0 = (size_t)b * TT + (size_t)c * CH;

  const float4* kg = (const float4*)(kq + t0 * DK);
  const float4* ag = (const float4*)(alpha + t0 * DK);
  const float4* vg = (const float4*)(v + t0 * DV);
  const float4* bg = (const float4*)(beta + t0);

#if USE_ASYNC_LDS
  // Direct global->LDS async copies (ASYNCcnt), one s_wait_asynccnt at the end.
#pragma unroll
  for (int j = 0; j < (CH * DK) / (32 * 4); ++j) {
    int i = lane + j * 32;
    async_cp_b128(kg + i, &((float4*)Kl)[i]);
    async_cp_b128(ag + i, &((float4*)Al)[i]);
  }
#pragma unroll
  for (int j = 0; j < (CH * DV) / (32 * 4); ++j) {
    int i = lane + j * 32;
    async_cp_b128(vg + i, &((float4*)Vl)[i]);
  }
  async_cp_b128(bg + lane, &((float4*)Bl)[lane]);
  __builtin_amdgcn_s_wait_asynccnt(0);
  __syncthreads();
#else
#pragma unroll
  for (int j = 0; j < (CH * DK) / (32 * 4); ++j) {
    int i = lane + j * 32;
    ((float4*)Kl)[i] = kg[i];
    ((float4*)Al)[i] = ag[i];
  }
#pragma unroll
  for (int j = 0; j < (CH * DV) / (32 * 4); ++j) {
    int i = lane + j * 32;
    ((float4*)Vl)[i] = vg[i];
  }
  ((float4*)Bl)[lane] = bg[lane];
  __syncthreads();
#endif

  float pcol[DK], ucol[DK];
#pragma unroll
  for (int d = 0; d < DK; ++d) {
    pcol[d] = (d == lane) ? 1.0f : 0.0f;
    ucol[d] = 0.0f;
  }

  for (int t = 0; t < CH; ++t) {
    const float k_own = Kl[t * DK + lane];
    const float a_own = Al[t * DK + lane];
    const float bt = Bl[t];
    const float vj = (lane < DV) ? Vl[t * DV + lane] : 0.0f;

    float sp = 0.0f, su = 0.0f;
#pragma unroll
    for (int d = 0; d < DK; ++d) {
      const float kd = lane_bcast(k_own, d);
      const float ad = lane_bcast(a_own, d);
      pcol[d] *= ad;
      sp = fmaf(kd, pcol[d], sp);
      ucol[d] *= ad;
      su = fmaf(kd, ucol[d], su);
    }
    const float cp = -bt * sp;
    const float cu = bt * (vj - su);
#pragma unroll
    for (int d = 0; d < DK; ++d) {
      const float kd = lane_bcast(k_own, d);
      pcol[d] = fmaf(cp, kd, pcol[d]);
      ucol[d] = fmaf(cu, kd, ucol[d]);
    }
  }

  // Write P row-major (lane owns column `lane`), U row-major 32x16 padded.
  float* Pb = wsP + (size_t)blockIdx.x * (DK * DK);
#pragma unroll
  for (int d = 0; d < DK; ++d) Pb[d * DK + lane] = pcol[d];
  float* Ub = wsU + (size_t)blockIdx.x * (DK * 16);
  if (lane < 16) {
#pragma unroll
    for (int d = 0; d < DK; ++d) Ub[d * 16 + lane] = ucol[d];
  }
}

// ---------------------------------------------------------------------------
// Phase 2: per batch, sequentially compose chunk states with FP32 WMMA:
//   S_{c+1} (32x16) = P_c (32x32) @ S_c (32x16) + U_c
// using V_WMMA_F32_16X16X4_F32: 2 M-tiles x 8 K-steps = 16 WMMA per chunk.
// State S lives in LDS row-major [row*16+col].
// ---------------------------------------------------------------------------
__global__ __launch_bounds__(32) void kda_phase2(
    const float* __restrict__ wsP, const float* __restrict__ wsU,
    float* __restrict__ wsS) {
  __shared__ float S[DK * 16];  // 32x16
  const int lane = threadIdx.x;
  const int b = blockIdx.x;
  const int hi = lane >> 4;   // lane half
  const int lo = lane & 15;   // M (for A) / N (for B,C,D)

#pragma unroll
  for (int i = 0; i < 16; ++i) S[lane + i * 32] = 0.0f;
  __syncthreads();

  for (int c = 0; c < NC; ++c) {
    const size_t blk = (size_t)b * NC + c;
    const float* Pb = wsP + blk * (DK * DK);
    const float* Ub = wsU + blk * (DK * 16);
    float* Sb = wsS + blk * (DK * 16);

    // Emit chunk-start state for phase 3 (linear copy of 512 floats).
#pragma unroll
    for (int i = 0; i < 16; ++i) Sb[lane + i * 32] = S[lane + i * 32];

    // C/D layout: VGPR r -> row (r + 8*hi), col lo. Init accumulators from U.
    v8f acc0, acc1;
#pragma unroll
    for (int r = 0; r < 8; ++r) acc0[r] = Ub[(r + 8 * hi) * 16 + lo];
#pragma unroll
    for (int r = 0; r < 8; ++r) acc1[r] = Ub[(16 + r + 8 * hi) * 16 + lo];

#pragma unroll
    for (int k8 = 0; k8 < 8; ++k8) {
      const int kk = 4 * k8 + 2 * hi;  // even -> 8B aligned A loads
      // B 4x16 tile: rows kk, kk+1 of S at column lo.
      v2f bb;
      bb[0] = S[kk * 16 + lo];
      bb[1] = S[(kk + 1) * 16 + lo];
      // A 16x4 tiles: row lo (tile m) of P, K = kk, kk+1.
      v2f a0 = *(const v2f*)(Pb + lo * DK + kk);
      v2f a1 = *(const v2f*)(Pb + (16 + lo) * DK + kk);
      acc0 = __builtin_amdgcn_wmma_f32_16x16x4_f32(
          false, a0, false, bb, (short)0, acc0, false, false);
      acc1 = __builtin_amdgcn_wmma_f32_16x16x4_f32(
          false, a1, false, bb, (short)0, acc1, false, false);
    }
    __syncthreads();
#pragma unroll
    for (int r = 0; r < 8; ++r) S[(r + 8 * hi) * 16 + lo] = acc0[r];
#pragma unroll
    for (int r = 0; r < 8; ++r) S[(16 + r + 8 * hi) * 16 + lo] = acc1[r];
    __syncthreads();
  }
}

// ---------------------------------------------------------------------------
// Phase 3: per (batch, chunk), replay recursion from stored chunk-start state
// and emit out_t = q_t . S_t. Lane e (<6) owns state column e.
// ---------------------------------------------------------------------------
__global__ __launch_bounds__(32) void kda_phase3(
    const float* __restrict__ q, const float* __restrict__ kq,
    const float* __restrict__ v, const float* __restrict__ alpha,
    const float* __restrict__ beta, const float* __restrict__ wsS,
    float* __restrict__ out) {
  __shared__ float Ql[CH * DK];
  __shared__ float Kl[CH * DK];
  __shared__ float Al[CH * DK];
  __shared__ float Vl[CH * DV];
  __shared__ float Bl[CH];

  const int lane = threadIdx.x;
  const int b = blockIdx.x / NC;
  const int c = blockIdx.x % NC;
  const size_t t0 = (size_t)b * TT + (size_t)c * CH;

  const float4* qg = (const float4*)(q + t0 * DK);
  const float4* kg = (const float4*)(kq + t0 * DK);
  const float4* ag = (const float4*)(alpha + t0 * DK);
  const float4* vg = (const float4*)(v + t0 * DV);
  const float4* bg = (const float4*)(beta + t0);

#if USE_ASYNC_LDS
#pragma unroll
  for (int j = 0; j < (CH * DK) / (32 * 4); ++j) {
    int i = lane + j * 32;
    async_cp_b128(qg + i, &((float4*)Ql)[i]);
    async_cp_b128(kg + i, &((float4*)Kl)[i]);
    async_cp_b128(ag + i, &((float4*)Al)[i]);
  }
#pragma unroll
  for (int j = 0; j < (CH * DV) / (32 * 4); ++j) {
    int i = lane + j * 32;
    async_cp_b128(vg + i, &((float4*)Vl)[i]);
  }
  async_cp_b128(bg + lane, &((float4*)Bl)[lane]);
  __builtin_amdgcn_s_wait_asynccnt(0);
  __syncthreads();
#else
#pragma unroll
  for (int j = 0; j < (CH * DK) / (32 * 4); ++j) {
    int i = lane + j * 32;
    ((float4*)Ql)[i] = qg[i];
    ((float4*)Kl)[i] = kg[i];
    ((float4*)Al)[i] = ag[i];
  }
#pragma unroll
  for (int j = 0; j < (CH * DV) / (32 * 4); ++j) {
    int i = lane + j * 32;
    ((float4*)Vl)[i] = vg[i];
  }
  ((float4*)Bl)[lane] = bg[lane];
  __syncthreads();
#endif

  // Load chunk-start state column (row-major 32x16 in wsS).
  float scol[DK];
  const float* Sb = wsS + (size_t)blockIdx.x * (DK * 16);
#pragma unroll
  for (int d = 0; d < DK; ++d) scol[d] = Sb[d * 16 + (lane & 15)];

  for (int t = 0; t < CH; ++t) {
    const float k_own = Kl[t * DK + lane];
    const float a_own = Al[t * DK + lane];
    const float q_own = Ql[t * DK + lane];
    const float bt = Bl[t];
    const float ve = (lane < DV) ? Vl[t * DV + lane] : 0.0f;

    float s = 0.0f;
#pragma unroll
    for (int d = 0; d < DK; ++d) {
      const float kd = lane_bcast(k_own, d);
      const float ad = lane_bcast(a_own, d);
      scol[d] *= ad;
      s = fmaf(kd, scol[d], s);
    }
    const float cu = bt * (ve - s);
    float o = 0.0f;
#pragma unroll
    for (int d = 0; d < DK; ++d) {
      const float kd = lane_bcast(k_own, d);
      const float qd = lane_bcast(q_own, d);
      scol[d] = fmaf(cu, kd, scol[d]);
      o = fmaf(qd, scol[d], o);
    }
    if (lane < DV) out[(t0 + t) * DV + lane] = o;
  }
}

// ---------------------------------------------------------------------------
// Router gating: softmax -> stable top-4 -> cumulative-prob threshold mask,
// scatter selected probs back to original positions. One thread per row.
// ---------------------------------------------------------------------------
__global__ void router_kernel(const float* __restrict__ logits,
                              float* __restrict__ gates, int nrows) {
  int r = blockIdx.x * blockDim.x + threadIdx.x;
  if (r >= nrows) return;
  const float* L = logits + (size_t)r * EE;

  float m = -3.402823466e38f;
#pragma unroll
  for (int e = 0; e < EE; ++e) m = fmaxf(m, L[e]);
  float s = 0.0f;
#pragma unroll
  for (int e = 0; e < EE; ++e) s += expf(L[e] - m);

  // Stable top-4 by logit (strict > keeps earlier index on ties, matching
  // jnp.argsort(-p) tie-breaking).
  float tv0 = -3.402823466e38f, tv1 = tv0, tv2 = tv0, tv3 = tv0;
  int ti0 = -1, ti1 = -1, ti2 = -1, ti3 = -1;
#pragma unroll
  for (int e = 0; e < EE; ++e) {
    float x = L[e];
    if (x > tv0) {
      tv3 = tv2; ti3 = ti2; tv2 = tv1; ti2 = ti1; tv1 = tv0; ti1 = ti0;
      tv0 = x; ti0 = e;
    } else if (x > tv1) {
      tv3 = tv2; ti3 = ti2; tv2 = tv1; ti2 = ti1; tv1 = x; ti1 = e;
    } else if (x > tv2) {
      tv3 = tv2; ti3 = ti2; tv2 = x; ti2 = e;
    } else if (x > tv3) {
      tv3 = x; ti3 = e;
    }
  }
  const float inv = 1.0f / s;
  float p0 = expf(tv0 - m) * inv;
  float p1 = expf(tv1 - m) * inv;
  float p2 = expf(tv2 - m) * inv;
  float p3 = expf(tv3 - m) * inv;

  // mask_i = (prefix_before_i < 0.8) with rank 0 forced on; max_k = 4.
  float pre = p0;
  bool s1 = pre < 0.8f; pre += p1;
  bool s2 = pre < 0.8f; pre += p2;
  bool s3 = pre < 0.8f;

  float* G = gates + (size_t)r * EE;
#pragma unroll
  for (int e = 0; e < EE; ++e) {
    float val = 0.0f;
    if (e == ti0) val = p0;
    if (s1 && e == ti1) val = p1;
    if (s2 && e == ti2) val = p2;
    if (s3 && e == ti3) val = p3;
    G[e] = val;
  }
}

// ---------------------------------------------------------------------------
extern "C" void kernel_launch(void* const* d_in, const int* in_sizes, int n_in,
                              void* d_out, int out_size, void* d_ws,
                              size_t ws_size, hipStream_t stream) {
  const float* q     = (const float*)d_in[0];
  const float* k     = (const float*)d_in[1];
  const float* v     = (const float*)d_in[2];
  const float* alpha = (const float*)d_in[3];
  const float* beta  = (const float*)d_in[4];
  const float* rlog  = (const float*)d_in[5];

  float* out   = (float*)d_out;                       // (B,T,DV)
  float* gates = out + (size_t)BB * TT * DV;          // (B,T,E)

  float* wsP = (float*)d_ws;                          // B*NC*32*32
  float* wsU = wsP + (size_t)BB * NC * DK * DK;       // B*NC*32*16
  float* wsS = wsU + (size_t)BB * NC * DK * 16;       // B*NC*32*16

  const int nrows = BB * TT;
  router_kernel<<<(nrows + 255) / 256, 256, 0, stream>>>(rlog, gates, nrows);
  kda_phase1<<<BB * NC, 32, 0, stream>>>(k, v, alpha, beta, wsP, wsU);
  kda_phase2<<<BB, 32, 0, stream>>>(wsP, wsU, wsS);
  kda_phase3<<<BB * NC, 32, 0, stream>>>(q, k, v, alpha, beta, wsS, out);
}